// CoupledCLEFOModel_2396591751766
// MI455X (gfx1250) — compile-verified
//
#include <hip/hip_runtime.h>

#define BATCH   262144
#define NDEP    16
#define KIND    32
#define EPSV    1e-7f
#define NITER   12
#define TILES   (BATCH / 16)
#define WPB     8           // waves per block
#define BLOCKS  (TILES / WPB)

typedef float v2f __attribute__((ext_vector_type(2)));
typedef float v8f __attribute__((ext_vector_type(8)));

// d_ws float layout: Minv[256] | MB[512] | MT[512] | mu[16]
#define WS_MINV 0
#define WS_MB   256
#define WS_MT   768
#define WS_MU   1280

// ---------------------------------------------------------------------------
// Prep: invert M = (1+eps)I - Gamma with wave-parallel Gauss-Jordan
// (column-per-lane, pivot broadcast via shfl; M is diagonally dominant so no
// pivoting), then fold: MB = Minv@Bmat, MT = Minv@Theta, mu = Minv@Upsilon.
// ---------------------------------------------------------------------------
__global__ __launch_bounds__(32) void clefo_prep(
    const float* __restrict__ Ups, const float* __restrict__ Bmat,
    const float* __restrict__ Theta, const float* __restrict__ Gamma,
    float* __restrict__ ws) {
  __shared__ float sminv[256];
  const int l = threadIdx.x;          // 0..31
  float a[16];
  if (l < 16) {                        // lanes 0..15: columns of M
    #pragma unroll
    for (int i = 0; i < 16; ++i)
      a[i] = ((i == l) ? (1.0f + EPSV) : 0.0f) - Gamma[i * 16 + l];
  } else {                             // lanes 16..31: columns of I
    const int j = l - 16;
    #pragma unroll
    for (int i = 0; i < 16; ++i) a[i] = (i == j) ? 1.0f : 0.0f;
  }
  // Gauss-Jordan over the augmented [M | I]
  for (int p = 0; p < 16; ++p) {
    const float piv = __shfl(a[p], p, 32);
    const float inv = 1.0f / piv;
    a[p] *= inv;
    #pragma unroll
    for (int r = 0; r < 16; ++r) {
      if (r == p) continue;
      const float f = __shfl(a[r], p, 32);   // element (r,p), pre-update
      a[r] -= f * a[p];
    }
  }
  if (l >= 16) {                       // lanes 16..31 now hold Minv columns
    const int j = l - 16;
    #pragma unroll
    for (int i = 0; i < 16; ++i) {
      sminv[i * 16 + j] = a[i];
      ws[WS_MINV + i * 16 + j] = a[i];
    }
  }
  __syncthreads();
  // MB = Minv @ Bmat, MT = Minv @ Theta  (lane k owns column k, k=0..31)
  for (int i = 0; i < 16; ++i) {
    float sb = 0.0f, st = 0.0f;
    #pragma unroll
    for (int m = 0; m < 16; ++m) {
      const float w = sminv[i * 16 + m];
      sb += w * Bmat[m * 32 + l];
      st += w * Theta[m * 32 + l];
    }
    ws[WS_MB + i * 32 + l] = sb;
    ws[WS_MT + i * 32 + l] = st;
  }
  if (l < 16) {                        // mu = Minv @ Upsilon[:,0]
    float s = 0.0f;
    #pragma unroll
    for (int m = 0; m < 16; ++m) s += sminv[l * 16 + m] * Ups[m];
    ws[WS_MU + l] = s;
  }
}

// ---------------------------------------------------------------------------
// Main: one wave solves a 16-item tile.
//   c = mu + x@MB^T + z@MT^T   (16 WMMA f32 16x16x4)
//   d = x@Lam^T                ( 8 WMMA)
//   y <- c + (d .* y)@Minv^T   ( 4 WMMA per iteration, NITER iterations)
// C/D layout (vgpr r, lane l) = element (row r+8*(l>>4), col l&15); rows=items.
// A-frag (K-slice k0): lane l, vgpr v = element (l&15, k0 + v + 2*(l>>4)).
// B-frag (K-slice k0): lane l, vgpr v = element (k0 + v + 2*(l>>4), l&15).
// ---------------------------------------------------------------------------
#define XPITCH 36   // LDS row pitch for 16x32 tiles: 16B-aligned, conflict-free
#define TPITCH 18   // LDS row pitch for 16x16 transpose buffer
#define WAVE_LDS (2 * 16 * XPITCH + 16 * TPITCH)   // 1440 floats / wave

__global__ __launch_bounds__(256) void clefo_solve(
    const float* __restrict__ X, const float* __restrict__ Z,
    const float* __restrict__ Lam, const float* __restrict__ ws,
    float* __restrict__ out) {
  __shared__ float lds[WPB * WAVE_LDS];
  const int wid  = threadIdx.x >> 5;
  const int lane = threadIdx.x & 31;
  const int hi   = lane >> 4;     // half-wave
  const int lo   = lane & 15;
  const int tile = blockIdx.x * WPB + wid;

  float* lx = lds + wid * WAVE_LDS;
  float* lz = lx + 16 * XPITCH;
  float* lt = lz + 16 * XPITCH;

  // ---- constant B-fragments (tiny, L2-resident) ----
  v2f fb[8], ft[8], fl[8], fm[4];
  #pragma unroll
  for (int s = 0; s < 8; ++s) {
    const int k = s * 4 + 2 * hi;
    fb[s] = *(const v2f*)(ws + WS_MB + lo * 32 + k);   // MB[n][k]
    ft[s] = *(const v2f*)(ws + WS_MT + lo * 32 + k);   // MT[n][k]
    fl[s] = *(const v2f*)(Lam + lo * 32 + k);          // Lam[n][k]
  }
  #pragma unroll
  for (int s = 0; s < 4; ++s)
    fm[s] = *(const v2f*)(ws + WS_MINV + lo * 16 + s * 4 + 2 * hi); // Minv[n][k]
  const float muv = ws[WS_MU + lo];

  // ---- stage X,Z tiles into padded LDS (coalesced b128) ----
  const float* Xg = X + (size_t)tile * (16 * KIND);
  const float* Zg = Z + (size_t)tile * (16 * KIND);
  #pragma unroll
  for (int i = 0; i < 4; ++i) {
    const int g = i * 128 + lane * 4;
    const int row = g >> 5, col = g & 31;
    *(float4*)(lx + row * XPITCH + col) = *(const float4*)(Xg + g);
    *(float4*)(lz + row * XPITCH + col) = *(const float4*)(Zg + g);
  }
  __syncthreads();

  // ---- c = mu + x@MB^T + z@MT^T ; d = x@Lam^T ----
  v8f c;
  #pragma unroll
  for (int r = 0; r < 8; ++r) c[r] = muv;
  #pragma unroll
  for (int s = 0; s < 8; ++s) {
    const v2f ax = *(const v2f*)(lx + lo * XPITCH + s * 4 + 2 * hi);
    c = __builtin_amdgcn_wmma_f32_16x16x4_f32(false, ax, false, fb[s],
                                              (short)0, c, false, false);
  }
  #pragma unroll
  for (int s = 0; s < 8; ++s) {
    const v2f az = *(const v2f*)(lz + lo * XPITCH + s * 4 + 2 * hi);
    c = __builtin_amdgcn_wmma_f32_16x16x4_f32(false, az, false, ft[s],
                                              (short)0, c, false, false);
  }
  v8f d = {0.f, 0.f, 0.f, 0.f, 0.f, 0.f, 0.f, 0.f};
  #pragma unroll
  for (int s = 0; s < 8; ++s) {
    const v2f ax = *(const v2f*)(lx + lo * XPITCH + s * 4 + 2 * hi);
    d = __builtin_amdgcn_wmma_f32_16x16x4_f32(false, ax, false, fl[s],
                                              (short)0, d, false, false);
  }

  // ---- fixed-point iterations: y <- c + (d .* y) @ Minv^T ----
  v8f y = c;
  #pragma unroll 2
  for (int it = 0; it < NITER; ++it) {
    #pragma unroll
    for (int r = 0; r < 8; ++r)
      lt[(r + 8 * hi) * TPITCH + lo] = d[r] * y[r];   // C-layout -> row-major
    v8f yn = c;
    #pragma unroll
    for (int s = 0; s < 4; ++s) {
      const v2f at = *(const v2f*)(lt + lo * TPITCH + s * 4 + 2 * hi); // A-frag
      yn = __builtin_amdgcn_wmma_f32_16x16x4_f32(false, at, false, fm[s],
                                                 (short)0, yn, false, false);
    }
    y = yn;
  }

  // ---- store tile (C-layout -> [B,16] row-major) ----
  float* og = out + (size_t)tile * (16 * NDEP);
  #pragma unroll
  for (int r = 0; r < 8; ++r) og[(r + 8 * hi) * NDEP + lo] = y[r];
}

// ---------------------------------------------------------------------------
extern "C" void kernel_launch(void* const* d_in, const int* in_sizes, int n_in,
                              void* d_out, int out_size, void* d_ws, size_t ws_size,
                              hipStream_t stream) {
  const float* X     = (const float*)d_in[0];
  const float* Z     = (const float*)d_in[1];
  const float* Ups   = (const float*)d_in[2];
  const float* Bmat  = (const float*)d_in[3];
  const float* Theta = (const float*)d_in[4];
  const float* Gamma = (const float*)d_in[5];
  const float* Lam   = (const float*)d_in[6];
  float* ws  = (float*)d_ws;
  float* out = (float*)d_out;

  clefo_prep<<<1, 32, 0, stream>>>(Ups, Bmat, Theta, Gamma, ws);
  clefo_solve<<<BLOCKS, 256, 0, stream>>>(X, Z, Lam, ws, out);
}